// BernNet_10273561772517
// MI455X (gfx1250) — compile-verified
//
#include <hip/hip_runtime.h>

typedef __attribute__((ext_vector_type(2))) float v2f;
typedef __attribute__((ext_vector_type(8))) float v8f;

#define NNODES 100000
#define NEDGES 1600000
#define HF 64

// ---------------- utility kernels ----------------

__global__ void zero_f32(float* __restrict__ p, int n) {
    int i = blockIdx.x * blockDim.x + threadIdx.x;
    if (i < n) p[i] = 0.0f;
}

__global__ void degree_scatter(const int* __restrict__ dst, float* __restrict__ deg, int E) {
    int e = blockIdx.x * blockDim.x + threadIdx.x;
    if (e < E) atomicAdd(&deg[dst[e]], 1.0f);
}

__global__ void make_dinv(float* __restrict__ d, int n) {
    int i = blockIdx.x * blockDim.x + threadIdx.x;
    if (i < n) {
        float v = d[i];
        v = v < 1.0f ? 1.0f : v;
        d[i] = rsqrtf(v);
    }
}

// ---------------- WMMA GEMM: out = relu(A(N x Kdim) @ W(Kdim x 64) + bias) ----------------
// 128 threads = 4 waves; block handles 16 rows, wave w handles cols [16w,16w+16).
// W staged in LDS (max 128*64 floats = 32 KB).
// f32 WMMA 16x16x4 lane layout (ISA 7.12.2):
//   A 16x4: lane m=lane&15; VGPR0/1 = K {0,1} for lanes 0-15, {2,3} for lanes 16-31.
//   B 4x16: mirrored (row k striped across lanes within a VGPR).
//   C/D 16x16: VGPR r -> row r (lanes 0-15) / row r+8 (lanes 16-31), col = lane&15.
__global__ __launch_bounds__(128)
void gemm_relu_wmma(const float* __restrict__ A, const float* __restrict__ W,
                    const float* __restrict__ bias, float* __restrict__ out, int Kdim) {
    __shared__ float sW[128 * HF];
    int tid = threadIdx.x;
    int total = Kdim * HF;
    for (int idx = tid; idx < total; idx += 128) sW[idx] = W[idx];
    __syncthreads();

    int lane = tid & 31;
    int wave = tid >> 5;
    int colb = wave * 16;
    int n    = lane & 15;
    int m    = lane & 15;
    int kb   = (lane >> 4) * 2;      // 0 for lanes 0-15, 2 for lanes 16-31

    long rowb = (long)blockIdx.x * 16;
    const float* arow = A + (size_t)(rowb + m) * Kdim + kb;

    v8f acc = {};
#pragma unroll 4
    for (int kk = 0; kk < Kdim; kk += 4) {
        v2f a = *(const v2f*)(arow + kk);
        v2f b;
        b[0] = sW[(kk + kb)     * HF + colb + n];
        b[1] = sW[(kk + kb + 1) * HF + colb + n];
        acc = __builtin_amdgcn_wmma_f32_16x16x4_f32(
            /*neg_a=*/false, a, /*neg_b=*/false, b,
            /*c_mod=*/(short)0, acc, /*reuse_a=*/false, /*reuse_b=*/false);
    }

    float bv = bias[colb + n];
    int mofs = (lane < 16) ? 0 : 8;
    float* obase = out + (size_t)(rowb + mofs) * HF + colb + n;
#pragma unroll
    for (int r = 0; r < 8; ++r) {
        float v = acc[r] + bv;
        obase[(size_t)r * HF] = v > 0.0f ? v : 0.0f;
    }
}

// ---------------- propagation scatter: agg[dst] += feat[src] * dinv[src] ----------------
// 16 threads per edge, 4 floats per thread (float4 gather + 4 f32 atomics).
__global__ void prop_scatter(const int* __restrict__ src, const int* __restrict__ dst,
                             const float* __restrict__ feat, const float* __restrict__ dinv,
                             float* __restrict__ agg, int E) {
    long t = (long)blockIdx.x * blockDim.x + threadIdx.x;
    long total = (long)E * 16;
    if (t >= total) return;
    int e = (int)(t >> 4);
    int j = ((int)t & 15) * 4;
    int s = src[e];
    int d = dst[e];
    float sc = dinv[s];
    float4 v = *(const float4*)(feat + (size_t)s * HF + j);
    float* ap = agg + (size_t)d * HF + j;
    atomicAdd(ap + 0, v.x * sc);
    atomicAdd(ap + 1, v.y * sc);
    atomicAdd(ap + 2, v.z * sc);
    atomicAdd(ap + 3, v.w * sc);
}

// ---------------- feat += agg * dinv (per-node scale), float4 granularity ----------------
__global__ void feat_update(float* __restrict__ feat, const float* __restrict__ agg,
                            const float* __restrict__ dinv, int n16) {
    int t = blockIdx.x * blockDim.x + threadIdx.x;   // over N*16 float4s
    if (t >= n16) return;
    int node = t >> 4;
    float sc = dinv[node];
    float4 fv = ((const float4*)feat)[t];
    float4 av = ((const float4*)agg)[t];
    fv.x += av.x * sc;
    fv.y += av.y * sc;
    fv.z += av.z * sc;
    fv.w += av.w * sc;
    ((float4*)feat)[t] = fv;
}

// ---------------- fused finalize: Lfeat, Bernstein combine, ReLU, 64->2 head ----------------
__global__ void finalize(const float* __restrict__ feat, const float* __restrict__ agg,
                         const float* __restrict__ dinv, const float* __restrict__ bern_w,
                         const float* __restrict__ Wm, const float* __restrict__ bm,
                         float* __restrict__ out, int n) {
    int i = blockIdx.x * blockDim.x + threadIdx.x;
    if (i >= n) return;
    float w0 = bern_w[0]; w0 = w0 > 0.0f ? w0 : 0.0f;
    float w1 = bern_w[1]; w1 = w1 > 0.0f ? w1 : 0.0f;
    float w2 = bern_w[2]; w2 = w2 > 0.0f ? w2 : 0.0f;
    const float c0 = 0.25f * w0;               // comb(2,0)/4 * w0 acting on feat
    const float c1 = 0.5f * w1 + 0.25f * w2;   // (comb(2,1)*w1 + comb(2,2)*w2)/4 on Lfeat
    float sc = dinv[i];
    const float* f = feat + (size_t)i * HF;
    const float* a = agg  + (size_t)i * HF;
    float o0 = 0.0f, o1 = 0.0f;
#pragma unroll 8
    for (int k = 0; k < HF; ++k) {
        float fk = f[k];
        float lf = fk - a[k] * sc;             // Lfeat = feat - prop(feat)
        float of = c0 * fk + c1 * lf;
        float h  = of > 0.0f ? of : 0.0f;
        o0 += h * Wm[k * 2 + 0];
        o1 += h * Wm[k * 2 + 1];
    }
    out[(size_t)i * 2 + 0] = o0 + bm[0];
    out[(size_t)i * 2 + 1] = o1 + bm[1];
}

// ---------------- host orchestration ----------------

extern "C" void kernel_launch(void* const* d_in, const int* in_sizes, int n_in,
                              void* d_out, int out_size, void* d_ws, size_t ws_size,
                              hipStream_t stream) {
    const float* feature = (const float*)d_in[0];
    const int*   src     = (const int*)  d_in[1];
    const int*   dst     = (const int*)  d_in[2];
    const float* W1      = (const float*)d_in[3];
    const float* b1      = (const float*)d_in[4];
    const float* W2      = (const float*)d_in[5];
    const float* b2      = (const float*)d_in[6];
    const float* bern_w  = (const float*)d_in[7];
    const float* Wm      = (const float*)d_in[8];
    const float* bm      = (const float*)d_in[9];
    float* outp = (float*)d_out;

    float* ws   = (float*)d_ws;
    float* feat = ws;                              // N*64
    float* tmpA = ws + (size_t)NNODES * HF;        // N*64 : h0, then agg scratch
    float* dinv = tmpA + (size_t)NNODES * HF;      // N

    const int B = 256;
    // degree -> dinv
    zero_f32<<<(NNODES + B - 1) / B, B, 0, stream>>>(dinv, NNODES);
    degree_scatter<<<(NEDGES + B - 1) / B, B, 0, stream>>>(dst, dinv, NEDGES);
    make_dinv<<<(NNODES + B - 1) / B, B, 0, stream>>>(dinv, NNODES);

    // MLP: h0 = relu(feature@W1+b1); feat = relu(h0@W2+b2)
    gemm_relu_wmma<<<NNODES / 16, 128, 0, stream>>>(feature, W1, b1, tmpA, 128);
    gemm_relu_wmma<<<NNODES / 16, 128, 0, stream>>>(tmpA, W2, b2, feat, HF);

    // K=2 propagation iterations: feat += dinv * scatter(feat*dinv)
    const int nv = NNODES * HF;
    for (int it = 0; it < 2; ++it) {
        zero_f32<<<(nv + B - 1) / B, B, 0, stream>>>(tmpA, nv);
        prop_scatter<<<(int)(((long)NEDGES * 16 + B - 1) / B), B, 0, stream>>>(
            src, dst, feat, dinv, tmpA, NEDGES);
        feat_update<<<(NNODES * 16 + B - 1) / B, B, 0, stream>>>(feat, tmpA, dinv, NNODES * 16);
    }

    // Laplacian pass feeding fused combine + head
    zero_f32<<<(nv + B - 1) / B, B, 0, stream>>>(tmpA, nv);
    prop_scatter<<<(int)(((long)NEDGES * 16 + B - 1) / B), B, 0, stream>>>(
        src, dst, feat, dinv, tmpA, NEDGES);
    finalize<<<(NNODES + B - 1) / B, B, 0, stream>>>(feat, tmpA, dinv, bern_w, Wm, bm,
                                                     outp, NNODES);
}